// STRNNCell_74783970558551
// MI455X (gfx1250) — compile-verified
//
#include <hip/hip_runtime.h>
#include <hip/hip_bf16.h>
#include <cmath>

#define BATCH   128
#define CTX     64
#define IN_SIZE 128
#define HID     128
#define NSLOT   9                     // NUM_SLOTS + 1
#define KTIME   (NSLOT * IN_SIZE)     // 1152
#define KTOT    (2 * KTIME + HID)     // 2432
#define KPAIRS  (KTOT / 2)            // 1216

typedef float v2f __attribute__((ext_vector_type(2)));
typedef float v8f __attribute__((ext_vector_type(8)));

// ---------------------------------------------------------------------------
// Phase 0: repack [TW; DW; HW] (K-major, KTOT x HID) into K-pair-interleaved
// layout: Bp[kp][n][p] = B[2*kp + p][n]  -> one b64 load per lane per K-chunk
// in the GEMM. One thread per (kp, n); float2 stores are fully coalesced.
// ---------------------------------------------------------------------------
__global__ void strnn_pack_b(const float* __restrict__ tw,
                             const float* __restrict__ dw,
                             const float* __restrict__ hw,
                             float* __restrict__ Bp) {
  const int tid = blockIdx.x * blockDim.x + threadIdx.x;   // < KPAIRS*HID
  const int kp  = tid / HID;
  const int n   = tid % HID;
  const int k   = 2 * kp;

  const float* src;
  int row = k;
  if (k < KTIME)            { src = tw; }
  else if (k < 2 * KTIME)   { src = dw; row = k - KTIME; }
  else                      { src = hw; row = k - 2 * KTIME; }

  v2f v;
  v.x = src[(size_t)row * HID + n];
  v.y = src[(size_t)(row + 1) * HID + n];
  *(v2f*)&Bp[(size_t)kp * (2 * HID) + 2 * n] = v;
}

// ---------------------------------------------------------------------------
// Phase 1: per-batch slot aggregation.
// Builds A in K-pair-interleaved, batch-column layout:
//   Ap[kp][b][p] = A[b][2*kp + p]   with the concatenated K axis
//     k in [0,1152)      : 0.5 * sum of masked x rows whose time slot == s
//     k in [1152,2304)   : 0.5 * sum of masked x rows whose dist slot == s
//     k in [2304,2432)   : h
// One block per batch, 128 threads (thread i owns feature i).
// ---------------------------------------------------------------------------
__global__ void strnn_aggregate(const float* __restrict__ x,
                                const int* __restrict__ tctx,
                                const int* __restrict__ dctx,
                                const unsigned char* __restrict__ mask,
                                const float* __restrict__ h,
                                float* __restrict__ Ap) {
  __shared__ float aggT[NSLOT][IN_SIZE];
  __shared__ float aggD[NSLOT][IN_SIZE];
  const int b = blockIdx.x;
  const int i = threadIdx.x;

#pragma unroll
  for (int s = 0; s < NSLOT; ++s) {
    aggT[s][i] = 0.0f;
    aggD[s][i] = 0.0f;
  }
  __syncthreads();

  const int base = b * CTX;
  for (int c = 0; c < CTX; ++c) {
    if (mask[base + c]) {
      const int   t  = tctx[base + c];   // uniform across the block
      const int   d  = dctx[base + c];
      const float xv = x[(size_t)(base + c) * IN_SIZE + i];
      aggT[t][i] += xv;                  // LDS indexed RMW, thread-private column
      aggD[d][i] += xv;
    }
  }
  __syncthreads();

  // scatter into pair-interleaved layout: k = (segment + i), kp = k>>1, p = k&1
  const int p = i & 1;
#pragma unroll
  for (int s = 0; s < NSLOT; ++s) {
    const int kt = s * IN_SIZE + i;            // time segment K index
    const int kd = KTIME + s * IN_SIZE + i;    // dist segment K index
    Ap[(size_t)(kt >> 1) * (2 * BATCH) + 2 * b + p] = 0.5f * aggT[s][i];
    Ap[(size_t)(kd >> 1) * (2 * BATCH) + 2 * b + p] = 0.5f * aggD[s][i];
  }
  const int kh = 2 * KTIME + i;
  Ap[(size_t)(kh >> 1) * (2 * BATCH) + 2 * b + p] = h[(size_t)b * HID + i];
}

// ---------------------------------------------------------------------------
// Phase 2: C[128x128] = A[128x2432] @ B[2432x128], then sigmoid.
// One wave32 per 16x16 output tile, K-chunks of 4 via V_WMMA_F32_16X16X4_F32.
// VGPR layouts (ISA 7.12.2, 32-bit):
//   A vgpr v : M = lane&15, K = 2*(lane>>4) + v
//   B vgpr v : N = lane&15, K = 2*(lane>>4) + v
//   D vgpr r : N = lane&15, M = r + 8*(lane>>4)
// Pair-interleaved staging => each lane grabs both K values with one b64 load.
// ---------------------------------------------------------------------------
__global__ void strnn_gemm_wmma(const float* __restrict__ Ap,
                                const float* __restrict__ Bp,
                                float* __restrict__ out) {
  const int tm   = blockIdx.x;          // batch tile 0..7
  const int tn   = blockIdx.y;          // hid   tile 0..7
  const int lane = threadIdx.x;         // 0..31
  const int half = lane >> 4;
  const int lr   = lane & 15;
  const int mcol = tm * 16 + lr;        // batch index
  const int ncol = tn * 16 + lr;        // hid index

  const float* ApL = Ap + 2 * mcol;     // lane-fixed column offset
  const float* BpL = Bp + 2 * ncol;

  v8f acc = {0.0f, 0.0f, 0.0f, 0.0f, 0.0f, 0.0f, 0.0f, 0.0f};

#pragma unroll 8
  for (int k = 0; k < KTOT; k += 4) {
    const int kp = (k >> 1) + half;     // pair row for this lane-half
    v2f a  = *(const v2f*)&ApL[(size_t)kp * (2 * BATCH)];
    v2f bb = *(const v2f*)&BpL[(size_t)kp * (2 * HID)];
    acc = __builtin_amdgcn_wmma_f32_16x16x4_f32(
        /*neg_a=*/false, a, /*neg_b=*/false, bb,
        /*c_mod=*/(short)0, acc, /*reuse_a=*/false, /*reuse_b=*/false);
  }

#pragma unroll
  for (int r = 0; r < 8; ++r) {
    const int row = tm * 16 + r + half * 8;
    // fast sigmoid: v_exp_f32 + v_rcp_f32
    const float e = __expf(-acc[r]);
    out[(size_t)row * HID + tn * 16 + lr] = __builtin_amdgcn_rcpf(1.0f + e);
  }
}

// ---------------------------------------------------------------------------
extern "C" void kernel_launch(void* const* d_in, const int* in_sizes, int n_in,
                              void* d_out, int out_size, void* d_ws, size_t ws_size,
                              hipStream_t stream) {
  const float*         x    = (const float*)d_in[0];
  const int*           tctx = (const int*)d_in[1];
  const int*           dctx = (const int*)d_in[2];
  const unsigned char* mask = (const unsigned char*)d_in[3];  // jnp.bool_ = 1 byte
  const float*         h    = (const float*)d_in[4];
  const float*         tw   = (const float*)d_in[5];
  const float*         dw   = (const float*)d_in[6];
  const float*         hw   = (const float*)d_in[7];
  float*               out  = (float*)d_out;

  char*  ws = (char*)d_ws;
  float* Ap = (float*)ws;                                         // KPAIRS x BATCH x 2
  float* Bp = (float*)(ws + (size_t)KTOT * BATCH * sizeof(float)); // KPAIRS x HID x 2

  // Phase 0: pack weights into pair-interleaved B
  {
    const int total = KPAIRS * HID;     // 155,648 threads
    strnn_pack_b<<<total / 256, 256, 0, stream>>>(tw, dw, hw, Bp);
  }

  // Phase 1: slot aggregation -> pair-interleaved A
  strnn_aggregate<<<BATCH, IN_SIZE, 0, stream>>>(x, tctx, dctx, mask, h, Ap);

  // Phase 2: WMMA GEMM + sigmoid
  dim3 grid(BATCH / 16, HID / 16);
  strnn_gemm_wmma<<<grid, 32, 0, stream>>>(Ap, Bp, out);
}